// APLoss_45655502356908
// MI455X (gfx1250) — compile-verified
//
#include <hip/hip_runtime.h>

typedef float v2f __attribute__((ext_vector_type(2)));
typedef float v4f __attribute__((ext_vector_type(4)));
typedef float v8f __attribute__((ext_vector_type(8)));

#define AP_MARGIN 1.0f
#define AP_GAMMA  0.99f

// ---------------------------------------------------------------------------
// Phase 1: one block per 16-row block of the (P x N) pairwise surrogate.
// 8 waves per block; wave w sweeps columns [w*N/8, (w+1)*N/8).
// Row reduction is done on the matrix pipe: V_WMMA_F32_16X16X4_F32 with an
// all-ones B matrix accumulates sum_k A[i][k] into every column of C/D.
// A second WMMA with mask-multiplied A accumulates the positive-masked sum.
// A-matrix 16x4 f32 layout: lane L holds row L%16; component c is K = 2*hi+c
// (hi = L>>4). C/D layout: lane 0 comps 0..7 = rows 0..7, lane 16 = rows 8..15.
// ---------------------------------------------------------------------------
__global__ __launch_bounds__(256) void aploss_phase1(
    const float* __restrict__ y_pred,    // (N)
    const float* __restrict__ y_true,    // (N), 0/1 floats (mask)
    const int*   __restrict__ index_p,   // (P) dataset indices of positives
    const int*   __restrict__ pos_idx,   // (P) batch indices of positives
    const float* __restrict__ u_all,     // (POS_LEN)
    const float* __restrict__ u_pos,     // (POS_LEN)
    float*       __restrict__ block_part,// (P/16) per-block partial loss sums
    int N)
{
    __shared__ float lds_all[8][16];
    __shared__ float lds_pos[8][16];
    __shared__ float lds_term[16];

    const int tid  = threadIdx.x;
    const int lane = tid & 31;
    const int wave = tid >> 5;
    const int r    = lane & 15;   // row within 16-row block
    const int hi   = lane >> 4;   // half-wave selects K in {2*hi, 2*hi+1}

    const int i0 = blockIdx.x * 16;

    // Gather this lane's positive score: f = y_pred[pos_idx[i0 + r]]
    const float f    = y_pred[pos_idx[i0 + r]];
    const float base = AP_MARGIN - f;  // sur = max(base + y_pred[j], 0)^2

    const int colsPerWave = N >> 3;    // N/8, N=16384 -> 2048
    const int j0 = wave * colsPerWave;
    const int j1 = j0 + colsPerWave;

    v2f bOnes; bOnes[0] = 1.0f; bOnes[1] = 1.0f;   // all-ones B (layout-free)
    v8f accA = {};   // accumulates s_all for rows of this block
    v8f accP = {};   // accumulates s_pos

    for (int j = j0; j < j1; j += 4) {
        // All lanes read the same 16B -> broadcast loads, L2/WGP$ resident.
        v4f yv = *(const v4f*)(y_pred + j);
        v4f mv = *(const v4f*)(y_true + j);
        float ya = hi ? yv[2] : yv[0];
        float yb = hi ? yv[3] : yv[1];
        float ma = hi ? mv[2] : mv[0];
        float mb = hi ? mv[3] : mv[1];

        float ta = fmaxf(base + ya, 0.0f);
        float tb = fmaxf(base + yb, 0.0f);
        float sa = ta * ta;
        float sb = tb * tb;

        v2f A;  A[0]  = sa;      A[1]  = sb;        // 16x4 sur tile
        v2f Ap; Ap[0] = sa * ma; Ap[1] = sb * mb;   // masked tile

        // D = A x ones(4x16) + C  ==> every column of D holds row sums.
        accA = __builtin_amdgcn_wmma_f32_16x16x4_f32(
            false, A,  false, bOnes, (short)0, accA, false, false);
        accP = __builtin_amdgcn_wmma_f32_16x16x4_f32(
            false, Ap, false, bOnes, (short)0, accP, false, false);
    }

    // Column N=0 lives in lane 0 (rows 0..7) and lane 16 (rows 8..15).
    if (lane == 0 || lane == 16) {
        const int rb = hi * 8;
        #pragma unroll
        for (int g = 0; g < 8; ++g) {
            lds_all[wave][rb + g] = accA[g];
            lds_pos[wave][rb + g] = accP[g];
        }
    }
    __syncthreads();

    // Combine the 8 waves' partial row sums, then the per-row loss term.
    if (tid < 16) {
        float sAll = 0.0f, sPos = 0.0f;
        #pragma unroll
        for (int w = 0; w < 8; ++w) {
            sAll += lds_all[w][tid];
            sPos += lds_pos[w][tid];
        }
        const float invN = 1.0f / (float)N;
        const int   i    = i0 + tid;
        const int   idx  = index_p[i];
        const float uaN  = (1.0f - AP_GAMMA) * u_all[idx] + AP_GAMMA * (sAll * invN);
        const float upN  = (1.0f - AP_GAMMA) * u_pos[idx] + AP_GAMMA * (sPos * invN);
        // sum_j p[i,j]*sur[i,j] = (upN*sAll - uaN*sPos) / uaN^2
        lds_term[tid] = (upN * sAll - uaN * sPos) / (uaN * uaN);
    }
    __syncthreads();

    if (tid == 0) {
        float s = 0.0f;
        #pragma unroll
        for (int t = 0; t < 16; ++t) s += lds_term[t];
        block_part[blockIdx.x] = s;
    }
}

// ---------------------------------------------------------------------------
// Phase 2: deterministic serial reduction of the 128 block partials.
// ---------------------------------------------------------------------------
__global__ void aploss_phase2(const float* __restrict__ block_part,
                              float* __restrict__ out,
                              int nblocks, float scale)
{
    if (blockIdx.x == 0 && threadIdx.x == 0) {
        float s = 0.0f;
        for (int i = 0; i < nblocks; ++i) s += block_part[i];
        out[0] = s * scale;
    }
}

extern "C" void kernel_launch(void* const* d_in, const int* in_sizes, int n_in,
                              void* d_out, int out_size, void* d_ws, size_t ws_size,
                              hipStream_t stream)
{
    const float* y_pred  = (const float*)d_in[0];
    const float* y_true  = (const float*)d_in[1];
    const int*   index_p = (const int*)  d_in[2];
    const int*   pos_idx = (const int*)  d_in[3];
    const float* u_all   = (const float*)d_in[4];
    const float* u_pos   = (const float*)d_in[5];

    const int N = in_sizes[0];   // 16384
    const int P = in_sizes[2];   // 2048

    float* part = (float*)d_ws;  // P/16 floats of scratch (512 B)
    const int nblocks = P / 16;  // 128

    aploss_phase1<<<nblocks, 256, 0, stream>>>(
        y_pred, y_true, index_p, pos_idx, u_all, u_pos, part, N);

    const float scale = 1.0f / ((float)P * (float)N);
    aploss_phase2<<<1, 32, 0, stream>>>(part, (float*)d_out, nblocks, scale);
}